// HDBLUT_13477607375180
// MI455X (gfx1250) — compile-verified
//
#include <hip/hip_runtime.h>

typedef __attribute__((ext_vector_type(4))) float v4f;
typedef __attribute__((ext_vector_type(2))) float v2f;

#define N_IMG 512
#define TILE  16
#define HALO  2
#define TDIM  (TILE + 2 * HALO) /* 20 */

typedef __attribute__((address_space(1))) int gint_t;
typedef __attribute__((address_space(3))) int lint_t;

__device__ __forceinline__ int iclamp(int v, int lo, int hi) {
  return v < lo ? lo : (v > hi ? hi : v);
}

// One thread per input pixel. Each thread:
//   - reads its 5x5 clamped neighborhood from an LDS tile,
//   - does 12 random 16B gathers (3 LUTs x 4 rotations) from the big tables,
//   - accumulates into a private 2x2 block with the per-rotation permutation,
//   - NT-stores the 2x2 output block (write-once stream, keep L2 for LUT gathers).
__global__ __launch_bounds__(256) void hdb_lut_kernel(
    const int* __restrict__ img,
    const float* __restrict__ wh,
    const float* __restrict__ wd,
    const float* __restrict__ wb,
    float* __restrict__ out)
{
  __shared__ int tile[TDIM][TDIM];

  const int bz  = blockIdx.z;
  const int by0 = blockIdx.y * TILE;
  const int bx0 = blockIdx.x * TILE;
  const int tid = threadIdx.y * TILE + threadIdx.x;

  const int* __restrict__ imgB = img + (size_t)bz * (N_IMG * N_IMG);

  // ---- LDS tile fill: CDNA5 async global->LDS path when the builtin exists ----
#if defined(__gfx1250__) && __has_builtin(__builtin_amdgcn_global_load_async_to_lds_b32)
  for (int e = tid; e < TDIM * TDIM; e += 256) {
    const int ty = e / TDIM, tx = e % TDIM;
    const int gy = iclamp(by0 + ty - HALO, 0, N_IMG - 1);
    const int gx = iclamp(bx0 + tx - HALO, 0, N_IMG - 1);
    __builtin_amdgcn_global_load_async_to_lds_b32(
        (gint_t*)(imgB + gy * N_IMG + gx),
        (lint_t*)&tile[ty][tx],
        /*offset=*/0, /*cpol=*/0);
  }
#if __has_builtin(__builtin_amdgcn_s_wait_asynccnt)
  __builtin_amdgcn_s_wait_asynccnt(0);
#else
  asm volatile("s_wait_asynccnt 0" ::: "memory");
#endif
#else
  for (int e = tid; e < TDIM * TDIM; e += 256) {
    const int ty = e / TDIM, tx = e % TDIM;
    const int gy = iclamp(by0 + ty - HALO, 0, N_IMG - 1);
    const int gx = iclamp(bx0 + tx - HALO, 0, N_IMG - 1);
    tile[ty][tx] = imgB[gy * N_IMG + gx];
  }
#endif
  __syncthreads();

  const int cy = threadIdx.y + HALO;
  const int cx = threadIdx.x + HALO;
  const unsigned a = (unsigned)tile[cy][cx];  // tap (0,0) is shared by all 12 passes

  float acc[4] = {0.f, 0.f, 0.f, 0.f};  // flat du*2+dv

  const float* const tabs[3] = {wh, wd, wb};
  // 2nd & 3rd tap offsets (dr,dc) per table; 1st tap is always (0,0)
  constexpr int OFF[3][2][2] = {
      {{0, 1}, {0, 2}},  // 'h'
      {{1, 1}, {2, 2}},  // 'd'
      {{1, 2}, {2, 1}},  // 'b'
  };
  // Where weight component k = u*2+v lands in the output 2x2 block, flat du*2+dv:
  // r0:(u,v)  r1:(v,1-u)  r2:(1-u,1-v)  r3:(1-v,u)   (derived & verified)
  constexpr int DEST[4][4] = {
      {0, 1, 2, 3},
      {1, 3, 0, 2},
      {3, 2, 1, 0},
      {2, 0, 3, 1},
  };

#pragma unroll
  for (int t = 0; t < 3; ++t) {
    const v4f* __restrict__ wt = (const v4f*)tabs[t];
#pragma unroll
    for (int r = 0; r < 4; ++r) {
      const int dr2 = OFF[t][0][0], dc2 = OFF[t][0][1];
      const int dr3 = OFF[t][1][0], dc3 = OFF[t][1][1];
      int dy2, dx2, dy3, dx3;
      if (r == 0)      { dy2 =  dr2; dx2 =  dc2; dy3 =  dr3; dx3 =  dc3; }
      else if (r == 1) { dy2 =  dc2; dx2 = -dr2; dy3 =  dc3; dx3 = -dr3; }
      else if (r == 2) { dy2 = -dr2; dx2 = -dc2; dy3 = -dr3; dx3 = -dc3; }
      else             { dy2 = -dc2; dx2 =  dr2; dy3 = -dc3; dx3 =  dr3; }
      const unsigned b = (unsigned)tile[cy + dy2][cx + dx2];
      const unsigned c = (unsigned)tile[cy + dy3][cx + dx3];
      const unsigned idx = (a << 16) | (b << 8) | c;   // a*65536 + b*256 + c
      const v4f w4 = wt[idx];  // default RT hint: we WANT L2 to cache LUT lines
      acc[DEST[r][0]] += w4.x;
      acc[DEST[r][1]] += w4.y;
      acc[DEST[r][2]] += w4.z;
      acc[DEST[r][3]] += w4.w;
    }
  }

  // ---- write the 2x2 output block (streaming, non-temporal) ----
  const int y = by0 + (int)threadIdx.y;
  const int x = bx0 + (int)threadIdx.x;
  float* o = out + ((size_t)bz * (2 * N_IMG) + 2 * y) * (2 * N_IMG) + 2 * x;
  const float s = (1.0f / 3.0f);
  v2f r0 = {acc[0] * s, acc[1] * s};
  v2f r1 = {acc[2] * s, acc[3] * s};
  __builtin_nontemporal_store(r0, (v2f*)o);
  __builtin_nontemporal_store(r1, (v2f*)(o + 2 * N_IMG));
}

extern "C" void kernel_launch(void* const* d_in, const int* in_sizes, int n_in,
                              void* d_out, int out_size, void* d_ws, size_t ws_size,
                              hipStream_t stream) {
  (void)in_sizes; (void)n_in; (void)out_size; (void)d_ws; (void)ws_size;
  const int*   img = (const int*)d_in[0];
  const float* wh  = (const float*)d_in[1];
  const float* wd  = (const float*)d_in[2];
  const float* wb  = (const float*)d_in[3];
  float* out = (float*)d_out;

  dim3 block(TILE, TILE, 1);                       // 256 threads = 8 wave32
  dim3 grid(N_IMG / TILE, N_IMG / TILE, 8);        // 32 x 32 x 8 blocks
  hdb_lut_kernel<<<grid, block, 0, stream>>>(img, wh, wd, wb, out);
}